// GCODLoss_4930622455851
// MI455X (gfx1250) — compile-verified
//
#include <hip/hip_runtime.h>
#include <hip/hip_bf16.h>
#include <math.h>

typedef __attribute__((ext_vector_type(2))) float v2f;
typedef __attribute__((ext_vector_type(4))) float v4f;
typedef __attribute__((ext_vector_type(8))) float v8f;

#define D_FEAT 48

// ---------------- init: zero degree array + accumulators ----------------
__global__ void k_init(unsigned* __restrict__ deg, double* __restrict__ acc,
                       int* __restrict__ maxb, int n) {
    int i = blockIdx.x * blockDim.x + threadIdx.x;
    if (i < n) deg[i] = 0u;
    if (i == 0) { acc[0] = 0.0; acc[1] = 0.0; *maxb = 0; }
}

// ---------------- per-node squared norm + batch max ----------------
__global__ void k_sqnorm(const float* __restrict__ x, const int* __restrict__ batch,
                         float* __restrict__ sq, int* __restrict__ maxb, int n) {
    int i = blockIdx.x * blockDim.x + threadIdx.x;
    __shared__ int smax[256];
    int b = 0;
    if (i < n) {
        const float* xr = x + (size_t)i * D_FEAT;
        float s = 0.f;
#pragma unroll
        for (int k = 0; k < D_FEAT; k += 4) {
            v4f v = *(const v4f*)(xr + k);
            s += v.x * v.x + v.y * v.y + v.z * v.z + v.w * v.w;
        }
        sq[i] = s;
        b = batch[i];
    }
    smax[threadIdx.x] = b;
    __syncthreads();
    for (int o = 128; o > 0; o >>= 1) {
        if ((int)threadIdx.x < o) smax[threadIdx.x] = max(smax[threadIdx.x], smax[threadIdx.x + o]);
        __syncthreads();
    }
    if (threadIdx.x == 0) atomicMax(maxb, smax[0]);
}

// ---------------- degree over valid (same-graph) edges, row side ----------------
__global__ void k_degree(const int* __restrict__ row, const int* __restrict__ col,
                         const int* __restrict__ batch, unsigned* __restrict__ deg, int E) {
    int e = blockIdx.x * blockDim.x + threadIdx.x;
    if (e < E) {
        int r = row[e], c = col[e];
        if (batch[r] == batch[c]) atomicAdd(&deg[r], 1u);
    }
}

// ---------------- deg^{-1/2} ----------------
__global__ void k_dinv(const unsigned* __restrict__ deg, float* __restrict__ dinv, int n) {
    int i = blockIdx.x * blockDim.x + threadIdx.x;
    if (i < n) {
        unsigned d = deg[i];
        dinv[i] = d ? rsqrtf((float)d) : 0.f;
    }
}

// ---------------- cross entropy (mean over graphs), single block ----------------
__global__ void k_ce(const float* __restrict__ logits, const int* __restrict__ labels,
                     double* __restrict__ acc, int G, int C) {
    __shared__ double red[256];
    double val = 0.0;
    for (int g = threadIdx.x; g < G; g += blockDim.x) {
        const float* lg = logits + (size_t)g * C;
        float m = lg[0];
        for (int c = 1; c < C; ++c) m = fmaxf(m, lg[c]);
        float s = 0.f;
        for (int c = 0; c < C; ++c) s += expf(lg[c] - m);
        float lp = lg[labels[g]] - m - logf(s);
        val += -(double)lp;
    }
    red[threadIdx.x] = val;
    __syncthreads();
    for (int o = blockDim.x >> 1; o > 0; o >>= 1) {
        if ((int)threadIdx.x < o) red[threadIdx.x] += red[threadIdx.x + o];
        __syncthreads();
    }
    if (threadIdx.x == 0) acc[1] = red[0] / (double)G;
}

// ---------------- edge energy: dots via V_WMMA_F32_16X16X4_F32 ----------------
// Tile = 16 edges. A[m][k] = x[row[e_m]][kb+k], B[k][n] = x[col[e_n]][kb+k].
// 12 chained K=4 WMMAs give C = A*B^T; diag(C)[m] = x_row[e_m] . x_col[e_m].
// lx2 = sq[r] + sq[c] - 2*dot. One f64 atomic per wave.
__global__ void k_edges(const float* __restrict__ x,
                        const int* __restrict__ row, const int* __restrict__ col,
                        const int* __restrict__ batch,
                        const float* __restrict__ sq, const float* __restrict__ dinv,
                        double* __restrict__ acc, int E) {
    const int lane   = threadIdx.x & 31;
    const int wave   = (int)((blockIdx.x * blockDim.x + threadIdx.x) >> 5);
    const int nwaves = (int)((gridDim.x * blockDim.x) >> 5);
    const int l16    = lane & 15;
    const int half   = lane >> 4;           // 0: K=0,1   1: K=2,3
    const int T      = (E + 15) >> 4;

    double lacc = 0.0;
    for (int t = wave; t < T; t += nwaves) {
        const int base = t << 4;
        int eL = base + l16; if (eL >= E) eL = E - 1;   // clamp, keep EXEC full
        const int rL = row[eL];
        const int cL = col[eL];
        const float* xr = x + (size_t)rL * D_FEAT + 2 * half;
        const float* xc = x + (size_t)cL * D_FEAT + 2 * half;

        v8f cacc = {0.f, 0.f, 0.f, 0.f, 0.f, 0.f, 0.f, 0.f};
#pragma unroll
        for (int kb = 0; kb < D_FEAT; kb += 4) {
            v2f a = *(const v2f*)(xr + kb);
            v2f b = *(const v2f*)(xc + kb);
            cacc = __builtin_amdgcn_wmma_f32_16x16x4_f32(
                false, a, false, b, (short)0, cacc, false, false);
        }

        // Diagonal: m<8 -> VGPR m @ lane m (N=lane); m>=8 -> VGPR m-8 @ lane m+16 (N=lane-16)
        float dot = 0.f;
#pragma unroll
        for (int j = 0; j < 8; ++j) {
            if (lane == j)      dot = cacc[j];
            if (lane == j + 24) dot = cacc[j];
        }
        bool own = (lane < 8) || (lane >= 24);
        int  m   = (lane < 16) ? lane : (lane - 16);
        int  e   = base + m;
        float term = 0.f;
        if (own && e < E) {
            int r = row[e], c = col[e];
            if (batch[r] == batch[c]) {
                float lx2 = sq[r] + sq[c] - 2.f * dot;
                term = dinv[r] * dinv[c] * lx2;
            }
        }
        lacc += (double)term;
    }
    for (int off = 16; off > 0; off >>= 1)
        lacc += __shfl_xor(lacc, off, 32);
    if (lane == 0) atomicAdd(acc, lacc);
}

// ---------------- combine ----------------
__global__ void k_combine(const double* __restrict__ acc, const int* __restrict__ maxb,
                          float* __restrict__ out) {
    if (threadIdx.x == 0 && blockIdx.x == 0) {
        double ng = (double)(*maxb) + 1.0;
        out[0] = (float)(acc[1] + acc[0] / ng);   // LAMBDA_SMOOTHNESS = 1.0
    }
}

extern "C" void kernel_launch(void* const* d_in, const int* in_sizes, int n_in,
                              void* d_out, int out_size, void* d_ws, size_t ws_size,
                              hipStream_t stream) {
    const float* logits = (const float*)d_in[0];
    const int*   labels = (const int*)d_in[1];
    const float* x      = (const float*)d_in[2];
    const int*   edge   = (const int*)d_in[3];
    const int*   batch  = (const int*)d_in[4];

    const int G = in_sizes[1];
    const int C = in_sizes[0] / (G > 0 ? G : 1);
    const int N = in_sizes[4];
    const int E = in_sizes[3] / 2;
    const int* row = edge;        // edge_index[0, :]
    const int* col = edge + E;    // edge_index[1, :]

    // workspace layout, 256B-aligned slabs of N*4 bytes
    char* ws = (char*)d_ws;
    size_t nb = (((size_t)N * 4) + 255) & ~(size_t)255;
    unsigned* deg  = (unsigned*)(ws);
    float*    dinv = (float*)(ws + nb);
    float*    sq   = (float*)(ws + 2 * nb);
    double*   acc  = (double*)(ws + 3 * nb);        // [0]=dirichlet sum, [1]=ce
    int*      maxb = (int*)(ws + 3 * nb + 64);

    const int tpb = 256;
    k_init  <<<(N + tpb - 1) / tpb, tpb, 0, stream>>>(deg, acc, maxb, N);
    k_sqnorm<<<(N + tpb - 1) / tpb, tpb, 0, stream>>>(x, batch, sq, maxb, N);
    k_degree<<<(E + tpb - 1) / tpb, tpb, 0, stream>>>(row, col, batch, deg, E);
    k_dinv  <<<(N + tpb - 1) / tpb, tpb, 0, stream>>>(deg, dinv, N);
    k_ce    <<<1, 256, 0, stream>>>(logits, labels, acc, G, C);
    k_edges <<<512, 256, 0, stream>>>(x, row, col, batch, sq, dinv, acc, E);
    k_combine<<<1, 1, 0, stream>>>(acc, maxb, (float*)d_out);
}